// RegionOfInterestAlignPyramid_60971355734783
// MI455X (gfx1250) — compile-verified
//
#include <hip/hip_runtime.h>
#include <math.h>

// RoI-Align over an FPN pyramid, CDNA5 (gfx1250) version.
//
// Bandwidth-bound op (~0.35 FLOP/byte -> ~11us at 23.3 TB/s HBM), so the
// kernel is built around the gfx1250 async LDS-DMA path:
//   - GLOBAL_LOAD_ASYNC_TO_LDS_B128 double-buffered gather (ASYNCcnt)
//   - s_wait_asynccnt / s_wait_dscnt split-counter pipelining
//   - wave32-friendly mapping: 64 lanes x float4 == one 256-ch corner vector
// WMMA is deliberately NOT used: every sample point gathers its own 4 corner
// vectors, so there is no shared matrix operand and tensor ops cannot reduce
// bytes moved.

typedef float v4f __attribute__((ext_vector_type(4)));

#define EXTENT   7
#define NPTS     49          // 7x7
#define CHANNELS 256
#define C4       64          // float4 per corner vector (256 ch)
#define PTS_PER_CHUNK 4      // 256 threads / 64 lanes-per-point
#define NCHUNKS  13          // ceil(49/4)

__device__ __forceinline__ void sample_coords(
    int pt, float ny1, float dny, float nx1, float dnx,
    float Hm1f, float Hm2f, int W,
    int &y0, int &x0, float &wy, float &wx)
{
  int py = pt / 7;
  int px = pt - py * 7;
  float gy = (float)py / 6.0f;               // reference: arange(7)/(extent-1)
  float gx = (float)px / 6.0f;
  float ys = (ny1 + dny * gy) * Hm1f;        // (H-1) scale
  float xs = (nx1 + dnx * gx) * Hm1f;        // W == H for these pyramids
  float y0f = fminf(fmaxf(floorf(ys), 0.0f), Hm2f);
  float x0f = fminf(fmaxf(floorf(xs), 0.0f), Hm2f);
  y0 = (int)y0f;
  x0 = (int)x0f;
  wy = ys - y0f;
  wx = xs - x0f;
  (void)W;
}

__global__ __launch_bounds__(256) void roi_align_pyramid_async(
    const float* __restrict__ metadata,
    const float* __restrict__ boxes,
    const float* __restrict__ p2, const float* __restrict__ p3,
    const float* __restrict__ p4, const float* __restrict__ p5,
    float* __restrict__ out,
    int H2, int H3, int H4, int H5)
{
  // [double-buffer][point-slot][corner][chan4] = 2*4*4*64*16B = 32 KB LDS
  __shared__ v4f sbuf[2][PTS_PER_CHUNK][4][C4];

  const int box  = blockIdx.x;
  const int tid  = threadIdx.x;
  const int ptl  = tid >> 6;     // point slot within chunk: 0..3
  const int lane = tid & 63;     // float4 channel index: 0..63

  const float rows = metadata[0];
  const float cols = metadata[1];

  const float x1 = boxes[box * 4 + 0];
  const float y1 = boxes[box * 4 + 1];
  const float x2 = boxes[box * 4 + 2];
  const float y2 = boxes[box * 4 + 3];

  // Level selection: min(5, max(2, 4 + round(log2(sqrt(h*w)/sqrt(area)))))
  const float bh = y2 - y1;
  const float bw = x2 - x1;
  const float roi = log2f(sqrtf(bh * bw) / sqrtf(rows * cols));
  const int level = (int)fminf(5.0f, fmaxf(2.0f, 4.0f + rintf(roi))); // rintf = half-to-even

  const float* feat; int H;
  if      (level == 2) { feat = p2; H = H2; }
  else if (level == 3) { feat = p3; H = H3; }
  else if (level == 4) { feat = p4; H = H4; }
  else                 { feat = p5; H = H5; }
  const int   W    = H;                 // square pyramid levels
  const float Hm1f = (float)(H - 1);
  const float Hm2f = (float)(H - 2);

  const float ny1 = y1 / (rows - 1.0f);
  const float dny = y2 / (rows - 1.0f) - ny1;
  const float nx1 = x1 / (cols - 1.0f);
  const float dnx = x2 / (cols - 1.0f) - nx1;

  const size_t rowstride = (size_t)W * CHANNELS;   // floats per feature row

  // Issue one chunk's gather (4 points x 4 corners x 1KB) as async LDS-DMA.
  // ISA note: the instruction offset is added to BOTH the global and LDS
  // addresses, so offset:1024 encodes the (x0+1) corner AND its LDS slot.
  auto issue = [&](int k) {
    int pt = k * PTS_PER_CHUNK + ptl;
    if (pt > NPTS - 1) pt = NPTS - 1;     // clamp tail: redundant valid loads,
                                          // keeps ASYNCcnt uniform per wave
    int y0, x0; float wy, wx;
    sample_coords(pt, ny1, dny, nx1, dnx, Hm1f, Hm2f, W, y0, x0, wy, wx);

    const float* a0 = feat + ((size_t)y0 * W + x0) * CHANNELS + (lane << 2);
    const float* a1 = a0 + rowstride;     // y0+1 row (x0 <= W-2 guaranteed)

    unsigned l0 = (unsigned)(uintptr_t)&sbuf[k & 1][ptl][0][lane]; // LDS byte offset
    unsigned l2 = l0 + 2 * C4 * 16;       // corners 2,3 region (+2048 B)

    asm volatile("global_load_async_to_lds_b128 %0, %1, off"
                 :: "v"(l0), "v"(a0) : "memory");                  // (y0  , x0  )
    asm volatile("global_load_async_to_lds_b128 %0, %1, off offset:1024"
                 :: "v"(l0), "v"(a0) : "memory");                  // (y0  , x0+1)
    asm volatile("global_load_async_to_lds_b128 %0, %1, off"
                 :: "v"(l2), "v"(a1) : "memory");                  // (y0+1, x0  )
    asm volatile("global_load_async_to_lds_b128 %0, %1, off offset:1024"
                 :: "v"(l2), "v"(a1) : "memory");                  // (y0+1, x0+1)
  };

  issue(0);                                // prime the pipeline

  for (int k = 0; k < NCHUNKS; ++k) {
    if (k + 1 < NCHUNKS) {
      // Before recycling the other LDS buffer, make sure our previous
      // ds_load reads of it have drained (DMA writes are unordered vs DS).
      asm volatile("s_wait_dscnt 0" ::: "memory");
      issue(k + 1);                        // prefetch next chunk
      // 8 async ops outstanding (chunk k + k+1); loads complete in order,
      // so <=4 means chunk k's data is resident in LDS.
      asm volatile("s_wait_asynccnt 4" ::: "memory");
    } else {
      asm volatile("s_wait_asynccnt 0" ::: "memory");
    }

    const int pt = k * PTS_PER_CHUNK + ptl;
    if (pt < NPTS) {
      int y0, x0; float wy, wx;
      sample_coords(pt, ny1, dny, nx1, dnx, Hm1f, Hm2f, W, y0, x0, wy, wx);

      const int b = k & 1;
      const v4f f00 = sbuf[b][ptl][0][lane];
      const v4f f01 = sbuf[b][ptl][1][lane];
      const v4f f10 = sbuf[b][ptl][2][lane];
      const v4f f11 = sbuf[b][ptl][3][lane];

      const float w00 = (1.0f - wy) * (1.0f - wx);
      const float w01 = (1.0f - wy) * wx;
      const float w10 = wy * (1.0f - wx);
      const float w11 = wy * wx;

      v4f o = f00 * w00 + f01 * w01 + f10 * w10 + f11 * w11;

      // Streamed output: write once, non-temporal to keep L2 for gathers.
      v4f* dst = (v4f*)(out + ((size_t)box * NPTS + pt) * CHANNELS) + lane;
      __builtin_nontemporal_store(o, dst);
    }
  }
}

static int level_dim_from_elems(int nelems) {
  // nelems = 1 * H * H * CHANNELS  ->  H
  int hw = nelems / CHANNELS;
  int h = (int)(sqrt((double)hw) + 0.5);
  return h;
}

extern "C" void kernel_launch(void* const* d_in, const int* in_sizes, int n_in,
                              void* d_out, int out_size, void* d_ws, size_t ws_size,
                              hipStream_t stream) {
  const float* metadata = (const float*)d_in[0];
  const float* boxes    = (const float*)d_in[1];
  const float* p2       = (const float*)d_in[2];
  const float* p3       = (const float*)d_in[3];
  const float* p4       = (const float*)d_in[4];
  const float* p5       = (const float*)d_in[5];
  float* out = (float*)d_out;

  const int nboxes = in_sizes[1] / 4;              // [1, N, 4]
  const int H2 = level_dim_from_elems(in_sizes[2]);
  const int H3 = level_dim_from_elems(in_sizes[3]);
  const int H4 = level_dim_from_elems(in_sizes[4]);
  const int H5 = level_dim_from_elems(in_sizes[5]);

  dim3 grid(nboxes), block(256);                   // 1 block (8 wave32) per box
  hipLaunchKernelGGL(roi_align_pyramid_async, grid, block, 0, stream,
                     metadata, boxes, p2, p3, p4, p5, out, H2, H3, H4, H5);

  (void)d_ws; (void)ws_size; (void)n_in; (void)out_size;
}